// EncoderLayer_21242908246380
// MI455X (gfx1250) — compile-verified
//
#include <hip/hip_runtime.h>
#include <math.h>

// ---------------------------------------------------------------------------
// EncoderLayer for MI455X (gfx1250, wave32).
// HBM-bound workload (369MB outputs vs ~157 GFLOP) -> use exact-fp32 WMMA
// (V_WMMA_F32_16X16X4_F32) for every matmul; fp32 softmax/LN/GELU elsewhere.
// ---------------------------------------------------------------------------

typedef float v2f __attribute__((ext_vector_type(2)));
typedef float v8f __attribute__((ext_vector_type(8)));

#define B_     8
#define K_     32
#define L_     256
#define D_     256
#define NHEAD_ 4
#define DK_    64
#define DINNER_ 1024
#define NTOK_  (B_ * K_ * L_)   // 65536 tokens
#define WAVES_ 8

__device__ __forceinline__ v8f wmma_f32(v2f a, v2f b, v8f c) {
  // D = A(16x4 f32) * B(4x16 f32) + C(16x16 f32); exact fp32 matrix op.
  return __builtin_amdgcn_wmma_f32_16x16x4_f32(false, a, false, b, (short)0, c,
                                               false, false);
}

__device__ __forceinline__ float gelu_exact(float x) {
  return 0.5f * x * (1.0f + erff(x * 0.70710678118654752440f));
}

// ---------------------------------------------------------------------------
// Generic row-major GEMM: C[M,N] = A[M,Kd] x B[Kd,N] (+bias)(+gelu)(+residual)
// One 16x16 output tile per wave; 8 waves per block stacked along M.
// EPI: 0=none, 1=+bias, 2=+bias+GELU, 3=+bias+residual
// ---------------------------------------------------------------------------
template <int EPI>
__global__ void gemm_wmma_kernel(const float* __restrict__ A,
                                 const float* __restrict__ B,
                                 const float* __restrict__ bias,
                                 const float* __restrict__ R,
                                 float* __restrict__ C, int N, int Kd) {
  const int lane = threadIdx.x;
  const int lm = lane & 15;      // row/col within half-wave
  const int lh = lane >> 4;      // half-wave select
  const int tn = blockIdx.x * 16;
  const int tm = (blockIdx.y * WAVES_ + threadIdx.y) * 16;

  const float* arow = A + (size_t)(tm + lm) * Kd;  // A: row m, cols k0+2*lh..+1
  const float* bp   = B + (size_t)(tn + lm);       // B: col n, rows strided by N

  v8f acc = {};
  for (int kk = 0; kk < Kd; kk += 4) {
    v2f a = *(const v2f*)(arow + kk + 2 * lh);
    v2f b;
    b[0] = bp[(size_t)(kk + 2 * lh) * N];
    b[1] = bp[(size_t)(kk + 2 * lh + 1) * N];
    acc = wmma_f32(a, b, acc);
  }

  const float bb = (EPI >= 1) ? bias[tn + lm] : 0.0f;
#pragma unroll
  for (int v = 0; v < 8; ++v) {
    const int row = tm + v + 8 * lh;   // C/D layout: VGPR v -> M=v+8*lh, N=lm
    float val = acc[v] + bb;
    if (EPI == 2) val = gelu_exact(val);
    if (EPI == 3) val += R[(size_t)row * N + tn + lm];
    C[(size_t)row * N + tn + lm] = val;
  }
}

// ---------------------------------------------------------------------------
// LayerNorm over d=256 per token (block = 1 token, 256 threads), optional
// residual add. var uses /d (jnp.var default).
// ---------------------------------------------------------------------------
__global__ void ln_kernel(const float* __restrict__ in,
                          const float* __restrict__ gamma,
                          const float* __restrict__ beta,
                          const float* __restrict__ res,
                          float* __restrict__ out, float eps) {
  __shared__ float sb[256];
  const int t = blockIdx.x, j = threadIdx.x;
  const size_t base = (size_t)t * D_;
  const float x = in[base + j];
  sb[j] = x;
  for (int off = 128; off > 0; off >>= 1) {
    __syncthreads();
    if (j < off) sb[j] += sb[j + off];
  }
  __syncthreads();
  const float m = sb[0] * (1.0f / D_);
  __syncthreads();
  const float d0 = x - m;
  sb[j] = d0 * d0;
  for (int off = 128; off > 0; off >>= 1) {
    __syncthreads();
    if (j < off) sb[j] += sb[j + off];
  }
  __syncthreads();
  const float var = sb[0] * (1.0f / D_);
  float y = d0 * rsqrtf(var + eps) * gamma[j] + beta[j];
  if (res) y += res[base + j];
  out[base + j] = y;
}

// ---------------------------------------------------------------------------
// Temporal attention scores: S[g, i, j] = (Q_g . K_g^T)/8, g = bk*4 + head,
// Q/K rows contiguous in token order. grid (L/16, L/128, B*K*NHEAD).
// ---------------------------------------------------------------------------
__global__ void scores_tem_kernel(const float* __restrict__ q,
                                  const float* __restrict__ kmat,
                                  float* __restrict__ s) {
  const int lane = threadIdx.x, lm = lane & 15, lh = lane >> 4;
  const int g = blockIdx.z;
  const int bk = g >> 2, h = g & 3;
  const int tn = blockIdx.x * 16;
  const int tm = (blockIdx.y * WAVES_ + threadIdx.y) * 16;
  const float* qb = q + ((size_t)bk * L_ + tm + lm) * D_ + h * DK_;
  const float* kb = kmat + ((size_t)bk * L_ + tn + lm) * D_ + h * DK_;
  v8f acc = {};
#pragma unroll
  for (int kk = 0; kk < DK_; kk += 4) {
    v2f a = *(const v2f*)(qb + kk + 2 * lh);
    v2f b = *(const v2f*)(kb + kk + 2 * lh);  // B[k][n] = K[n][k] -> contiguous
    acc = wmma_f32(a, b, acc);
  }
  float* o = s + (size_t)g * L_ * L_;
#pragma unroll
  for (int v = 0; v < 8; ++v)
    o[(size_t)(tm + v + 8 * lh) * L_ + tn + lm] = acc[v] * 0.125f;
}

// ---------------------------------------------------------------------------
// Temporal P*V: ctx[token, h*64+n] = sum_j P[g,i,j] V[bk*L+j, h*64+n].
// grid (DK_/16, L/128, B*K*NHEAD).
// ---------------------------------------------------------------------------
__global__ void pv_tem_kernel(const float* __restrict__ p,
                              const float* __restrict__ v,
                              float* __restrict__ ctx) {
  const int lane = threadIdx.x, lm = lane & 15, lh = lane >> 4;
  const int g = blockIdx.z;
  const int bk = g >> 2, h = g & 3;
  const int tn = blockIdx.x * 16;
  const int tm = (blockIdx.y * WAVES_ + threadIdx.y) * 16;
  const float* pb = p + (size_t)g * L_ * L_ + (size_t)(tm + lm) * L_;
  const float* vb = v + (size_t)bk * L_ * D_ + h * DK_ + tn + lm;
  v8f acc = {};
  for (int kk = 0; kk < L_; kk += 4) {
    v2f a = *(const v2f*)(pb + kk + 2 * lh);
    v2f b;
    b[0] = vb[(size_t)(kk + 2 * lh) * D_];
    b[1] = vb[(size_t)(kk + 2 * lh + 1) * D_];
    acc = wmma_f32(a, b, acc);
  }
#pragma unroll
  for (int vv = 0; vv < 8; ++vv)
    ctx[((size_t)bk * L_ + tm + vv + 8 * lh) * D_ + h * DK_ + tn + lm] = acc[vv];
}

// ---------------------------------------------------------------------------
// Type attention scores over K=32 (fixed b,l; tokens strided by L_*D_):
// S is [b,L,n,32,32]. grid.x = B*L*NHEAD (8192), block (32,4): wave -> tile.
// ---------------------------------------------------------------------------
__global__ void scores_type_kernel(const float* __restrict__ q,
                                   const float* __restrict__ kmat,
                                   float* __restrict__ s) {
  const int lane = threadIdx.x, lm = lane & 15, lh = lane >> 4;
  const int w = threadIdx.y;
  const int mt = (w >> 1) * 16, nt = (w & 1) * 16;
  const int g = blockIdx.x;
  const int h = g & 3, l = (g >> 2) & 255, b = g >> 10;
  const size_t tokm = (size_t)(b * K_ + mt + lm) * L_ + l;
  const size_t tokn = (size_t)(b * K_ + nt + lm) * L_ + l;
  const float* qb = q + tokm * D_ + h * DK_;
  const float* kb = kmat + tokn * D_ + h * DK_;
  v8f acc = {};
#pragma unroll
  for (int kk = 0; kk < DK_; kk += 4) {
    v2f a = *(const v2f*)(qb + kk + 2 * lh);
    v2f bv = *(const v2f*)(kb + kk + 2 * lh);
    acc = wmma_f32(a, bv, acc);
  }
  float* o = s + (size_t)g * (K_ * K_);
#pragma unroll
  for (int v = 0; v < 8; ++v)
    o[(mt + v + 8 * lh) * K_ + nt + lm] = acc[v] * 0.125f;
}

// ---------------------------------------------------------------------------
// Type P*V: [32,32]x[32,64] per (b,l,head); block (32,8): 8 tiles = full out.
// ---------------------------------------------------------------------------
__global__ void pv_type_kernel(const float* __restrict__ p,
                               const float* __restrict__ v,
                               float* __restrict__ ctx) {
  const int lane = threadIdx.x, lm = lane & 15, lh = lane >> 4;
  const int w = threadIdx.y;
  const int mt = (w >> 2) * 16, nt = (w & 3) * 16;
  const int g = blockIdx.x;
  const int h = g & 3, l = (g >> 2) & 255, b = g >> 10;
  const float* pb = p + (size_t)g * (K_ * K_) + (size_t)(mt + lm) * K_;
  v8f acc = {};
#pragma unroll
  for (int kk = 0; kk < K_; kk += 4) {
    v2f a = *(const v2f*)(pb + kk + 2 * lh);
    v2f bv;
    bv[0] = v[((size_t)(b * K_ + kk + 2 * lh) * L_ + l) * D_ + h * DK_ + nt + lm];
    bv[1] = v[((size_t)(b * K_ + kk + 2 * lh + 1) * L_ + l) * D_ + h * DK_ + nt + lm];
    acc = wmma_f32(a, bv, acc);
  }
#pragma unroll
  for (int vv = 0; vv < 8; ++vv)
    ctx[((size_t)(b * K_ + mt + vv + 8 * lh) * L_ + l) * D_ + h * DK_ + nt + lm] =
        acc[vv];
}

// ---------------------------------------------------------------------------
// Masked softmax, temporal: one row (len 256) per block. Mask then softmax,
// exactly like the reference (-1e4 fill, fully-masked rows -> uniform).
// ---------------------------------------------------------------------------
__global__ void softmax_tem_kernel(float* __restrict__ s,
                                   const int* __restrict__ mask) {
  __shared__ float sb[256];
  const int r = blockIdx.x;    // ((bk*4 + h) * L + i)
  const int j = threadIdx.x;
  const int i = r & 255;
  const int bk = r >> 10;
  const int mi = mask[bk * L_ + i];
  const int mj = mask[bk * L_ + j];
  const size_t base = (size_t)r * L_;
  float v = s[base + j];
  if (!(mi && mj)) v = -10000.0f;
  sb[j] = v;
  for (int off = 128; off > 0; off >>= 1) {
    __syncthreads();
    if (j < off) sb[j] = fmaxf(sb[j], sb[j + off]);
  }
  __syncthreads();
  const float mx = sb[0];
  __syncthreads();
  const float e = expf(v - mx);
  sb[j] = e;
  for (int off = 128; off > 0; off >>= 1) {
    __syncthreads();
    if (j < off) sb[j] += sb[j + off];
  }
  __syncthreads();
  s[base + j] = e / sb[0];
}

// ---------------------------------------------------------------------------
// Masked softmax, type: one row (len 32) per wave, wave32 shuffle reductions.
// ---------------------------------------------------------------------------
__global__ void softmax_type_kernel(float* __restrict__ s,
                                    const int* __restrict__ mask) {
  const int r = blockIdx.x * 8 + threadIdx.y;  // (((b*L + l)*4 + h)*32 + i)
  const int j = threadIdx.x;
  const int i = r & 31;
  const int l = (r >> 7) & 255;
  const int b = r >> 15;
  const int mi = mask[(b * K_ + i) * L_ + l];
  const int mj = mask[(b * K_ + j) * L_ + l];
  const size_t base = (size_t)r * K_;
  float v = s[base + j];
  if (!(mi && mj)) v = -10000.0f;
  float mx = v;
  for (int off = 16; off > 0; off >>= 1) mx = fmaxf(mx, __shfl_xor(mx, off, 32));
  const float e = expf(v - mx);
  float sum = e;
  for (int off = 16; off > 0; off >>= 1) sum += __shfl_xor(sum, off, 32);
  s[base + j] = e / sum;
}

// ---------------------------------------------------------------------------
static void launch_gemm(int epi, const float* A, const float* Bm,
                        const float* bias, const float* R, float* C, int M,
                        int N, int Kd, hipStream_t st) {
  dim3 g(N / 16, M / (16 * WAVES_)), b(32, WAVES_);
  switch (epi) {
    case 0: gemm_wmma_kernel<0><<<g, b, 0, st>>>(A, Bm, bias, R, C, N, Kd); break;
    case 1: gemm_wmma_kernel<1><<<g, b, 0, st>>>(A, Bm, bias, R, C, N, Kd); break;
    case 2: gemm_wmma_kernel<2><<<g, b, 0, st>>>(A, Bm, bias, R, C, N, Kd); break;
    default: gemm_wmma_kernel<3><<<g, b, 0, st>>>(A, Bm, bias, R, C, N, Kd); break;
  }
}

extern "C" void kernel_launch(void* const* d_in, const int* in_sizes, int n_in,
                              void* d_out, int out_size, void* d_ws,
                              size_t ws_size, hipStream_t stream) {
  const float* x    = (const float*)d_in[0];
  const int*   mask = (const int*)d_in[1];
  const float* Wq_t = (const float*)d_in[2];
  const float* Wk_t = (const float*)d_in[3];
  const float* Wv_t = (const float*)d_in[4];
  const float* fcw_t = (const float*)d_in[5];
  const float* fcb_t = (const float*)d_in[6];
  const float* lng_t = (const float*)d_in[7];
  const float* lnb_t = (const float*)d_in[8];
  const float* Wq_y = (const float*)d_in[9];
  const float* Wk_y = (const float*)d_in[10];
  const float* Wv_y = (const float*)d_in[11];
  const float* fcw_y = (const float*)d_in[12];
  const float* fcb_y = (const float*)d_in[13];
  const float* lng_y = (const float*)d_in[14];
  const float* lnb_y = (const float*)d_in[15];
  const float* w1 = (const float*)d_in[16];
  const float* b1 = (const float*)d_in[17];
  const float* w2 = (const float*)d_in[18];
  const float* b2 = (const float*)d_in[19];
  const float* lng = (const float*)d_in[20];
  const float* lnb = (const float*)d_in[21];

  // d_out = [out (T*D) | attn_tem (B*K*NH*L*L) | attn_type (B*L*NH*K*K)]
  float* out_p     = (float*)d_out;
  float* attn_tem  = out_p + (size_t)NTOK_ * D_;             // 16,777,216
  float* attn_type = attn_tem + (size_t)B_ * K_ * NHEAD_ * L_ * L_;  // +67,108,864

  // Workspace: 5 token-tensor slots of T*D floats + FFN hidden chunk.
  const size_t TD = (size_t)NTOK_ * D_;
  float* S0 = (float*)d_ws;
  float* S1 = S0 + TD;
  float* S2 = S1 + TD;
  float* S3 = S2 + TD;
  float* S4 = S3 + TD;
  float* H  = S4 + TD;  // 8192 x 1024 chunk

  const dim3 lnGrid(NTOK_), lnBlk(256);

  // ---- temporal self-attention ------------------------------------------
  ln_kernel<<<lnGrid, lnBlk, 0, stream>>>(x, lng, lnb, nullptr, S0, 1e-6f);
  launch_gemm(0, S0, Wq_t, nullptr, nullptr, S1, NTOK_, D_, D_, stream);
  launch_gemm(0, S0, Wk_t, nullptr, nullptr, S2, NTOK_, D_, D_, stream);
  launch_gemm(0, S0, Wv_t, nullptr, nullptr, S3, NTOK_, D_, D_, stream);

  scores_tem_kernel<<<dim3(L_ / 16, L_ / 128, B_ * K_ * NHEAD_),
                      dim3(32, WAVES_), 0, stream>>>(S1, S2, attn_tem);
  softmax_tem_kernel<<<dim3(B_ * K_ * NHEAD_ * L_), dim3(256), 0, stream>>>(
      attn_tem, mask);
  pv_tem_kernel<<<dim3(DK_ / 16, L_ / 128, B_ * K_ * NHEAD_), dim3(32, WAVES_),
                  0, stream>>>(attn_tem, S3, S0);

  launch_gemm(1, S0, fcw_t, fcb_t, nullptr, S1, NTOK_, D_, D_, stream);
  // tem_out = LN(proj, lng_t, eps=1e-5) + x   -> S2 (persists)
  ln_kernel<<<lnGrid, lnBlk, 0, stream>>>(S1, lng_t, lnb_t, x, S2, 1e-5f);

  // ---- type self-attention ----------------------------------------------
  ln_kernel<<<lnGrid, lnBlk, 0, stream>>>(S2, lng, lnb, nullptr, S0, 1e-6f);
  launch_gemm(0, S0, Wq_y, nullptr, nullptr, S1, NTOK_, D_, D_, stream);
  launch_gemm(0, S0, Wk_y, nullptr, nullptr, S3, NTOK_, D_, D_, stream);
  launch_gemm(0, S0, Wv_y, nullptr, nullptr, S4, NTOK_, D_, D_, stream);

  scores_type_kernel<<<dim3(B_ * L_ * NHEAD_), dim3(32, 4), 0, stream>>>(
      S1, S3, attn_type);
  softmax_type_kernel<<<dim3(B_ * L_ * NHEAD_ * K_ / 8), dim3(32, 8), 0,
                        stream>>>(attn_type, mask);
  pv_type_kernel<<<dim3(B_ * L_ * NHEAD_), dim3(32, 8), 0, stream>>>(
      attn_type, S4, S0);

  launch_gemm(1, S0, fcw_y, fcb_y, nullptr, S1, NTOK_, D_, D_, stream);
  // enc = LN(proj, lng_y, eps=1e-5) + tem_out  -> S3 (persists)
  ln_kernel<<<lnGrid, lnBlk, 0, stream>>>(S1, lng_y, lnb_y, S2, S3, 1e-5f);

  // ---- FFN (chunked hidden to bound workspace) --------------------------
  ln_kernel<<<lnGrid, lnBlk, 0, stream>>>(S3, lng, lnb, nullptr, S0, 1e-6f);
  const int CH = 8192;
  for (int c = 0; c < NTOK_ / CH; ++c) {
    const size_t ro = (size_t)c * CH * D_;
    launch_gemm(2, S0 + ro, w1, b1, nullptr, H, CH, DINNER_, D_, stream);
    launch_gemm(3, H, w2, b2, S3 + ro, S1 + ro, CH, D_, DINNER_, stream);
  }
  // final LN -> d_out (token order == [b,k,l,d] after the reference's swaps)
  ln_kernel<<<lnGrid, lnBlk, 0, stream>>>(S1, lng, lnb, nullptr, out_p, 1e-6f);
}